// GCN_27255862460802
// MI455X (gfx1250) — compile-verified
//
#include <hip/hip_runtime.h>
#include <math.h>

#define N_NODES 40000
#define N_EDGES 640000
#define F_IN    33
#define H       128
#define C_OUT   3
#define L_LAYERS 8
#define ALPHA   0.1f

typedef __attribute__((ext_vector_type(2))) float v2f;
typedef __attribute__((ext_vector_type(8))) float v8f;

// ---------------- input layer: x0 = relu(x @ W_in^T + b_in) ----------------
__global__ void gcn_input_kernel(const float* __restrict__ x,
                                 const float* __restrict__ W_in,
                                 const float* __restrict__ b_in,
                                 float* __restrict__ x0,
                                 float* __restrict__ h0) {
    int t = blockIdx.x * blockDim.x + threadIdx.x;
    if (t >= N_NODES * H) return;
    int row = t >> 7;          // / H
    int hc  = t & (H - 1);
    const float* xr = x + row * F_IN;
    const float* wr = W_in + hc * F_IN;
    float acc = b_in[hc];
#pragma unroll
    for (int k = 0; k < F_IN; ++k) acc = fmaf(xr[k], wr[k], acc);
    acc = fmaxf(acc, 0.0f);
    x0[t] = acc;
    h0[t] = acc;
}

// ---------------- xx = ALPHA * x0 (also clears scratch poison) ----------------
__global__ void gcn_init_xx(const float* __restrict__ x0, float* __restrict__ xx) {
    int t = blockIdx.x * blockDim.x + threadIdx.x;
    if (t >= N_NODES * H / 4) return;
    float4 v = ((const float4*)x0)[t];
    v.x *= ALPHA; v.y *= ALPHA; v.z *= ALPHA; v.w *= ALPHA;
    ((float4*)xx)[t] = v;
}

// ---------------- SpMM scatter: xx[row] += (1-ALPHA)*w[e]*h[col] ----------------
// One wave per edge: 32 lanes x float4 = 128 features. Gathers hit the 192MB L2
// (feature matrix is 20.5MB); float adds resolve at the L2 atomic units.
__global__ void gcn_spmm(const int* __restrict__ erow,
                         const int* __restrict__ ecol,
                         const float* __restrict__ ew,
                         const float* __restrict__ h,
                         float* __restrict__ xx) {
    int t = blockIdx.x * blockDim.x + threadIdx.x;
    int e = t >> 5;
    int lane = t & 31;
    if (e >= N_EDGES) return;
    int r = erow[e];
    int c = ecol[e];
    float s = (1.0f - ALPHA) * ew[e];
    float4 hv = *(const float4*)(h + (size_t)c * H + lane * 4);
    float* dst = xx + (size_t)r * H + lane * 4;
    __hip_atomic_fetch_add(dst + 0, s * hv.x, __ATOMIC_RELAXED, __HIP_MEMORY_SCOPE_AGENT);
    __hip_atomic_fetch_add(dst + 1, s * hv.y, __ATOMIC_RELAXED, __HIP_MEMORY_SCOPE_AGENT);
    __hip_atomic_fetch_add(dst + 2, s * hv.z, __ATOMIC_RELAXED, __HIP_MEMORY_SCOPE_AGENT);
    __hip_atomic_fetch_add(dst + 3, s * hv.w, __ATOMIC_RELAXED, __HIP_MEMORY_SCOPE_AGENT);
}

// ---------------- WMMA layer: h_out = h_in + relu((1-b)*xx + b*(xx@W)) ----------------
// Wave computes one 16x16 output tile; K=128 as 32 steps of V_WMMA_F32_16X16X4_F32.
// Block = 256 threads = 8 waves = 8 column tiles (full H=128 row of tiles).
__global__ __launch_bounds__(256)
void gcn_layer_wmma(const float* __restrict__ xx,
                    const float* __restrict__ W,
                    const float* __restrict__ h_in,
                    float* __restrict__ h_out,
                    float beta) {
    const int lane = threadIdx.x & 31;
    const int wave = threadIdx.x >> 5;      // column tile 0..7
    const int row0 = blockIdx.x * 16;
    const int col0 = wave * 16;
    const int m  = lane & 15;
    const int hi = lane >> 4;               // 0: K=0,1  1: K=2,3

    v8f c = {};
    // A fragment: A[m][kb + 2*hi + {0,1}]  (16x4 f32 layout, ISA 7.12.2)
    const float* arow = xx + (size_t)(row0 + m) * H + 2 * hi;
    // B fragment: B[kb + 2*hi + {0,1}][col0 + n], n = lane&15
    const float* bcol = W + (size_t)(2 * hi) * H + col0 + m;

#pragma unroll 8
    for (int kb = 0; kb < H; kb += 4) {
        v2f a = *(const v2f*)(arow + kb);   // 8B-aligned contiguous pair
        v2f b;
        b.x = bcol[(size_t)kb * H];
        b.y = bcol[(size_t)kb * H + H];
        c = __builtin_amdgcn_wmma_f32_16x16x4_f32(
                /*neg_a=*/false, a, /*neg_b=*/false, b,
                /*c_mod=*/(short)0, c, /*reuse_a=*/false, /*reuse_b=*/false);
    }

    const float omb = 1.0f - beta;
#pragma unroll
    for (int r = 0; r < 8; ++r) {
        // C/D layout: lane 0-15 -> M=r, lane 16-31 -> M=r+8; N = lane&15
        size_t idx = (size_t)(row0 + r + 8 * hi) * H + col0 + m;
        float xxv  = xx[idx];
        float outv = omb * xxv + beta * c[r];
        h_out[idx] = h_in[idx] + fmaxf(outv, 0.0f);
    }
}

// ---------------- output layer: out = h @ W_out^T + b_out ----------------
__global__ void gcn_output_kernel(const float* __restrict__ h,
                                  const float* __restrict__ W_out,
                                  const float* __restrict__ b_out,
                                  float* __restrict__ out) {
    int t = blockIdx.x * blockDim.x + threadIdx.x;
    if (t >= N_NODES * C_OUT) return;
    int row = t / C_OUT;
    int cc  = t - row * C_OUT;
    const float* hr = h + (size_t)row * H;
    const float* wr = W_out + cc * H;
    float acc = b_out[cc];
#pragma unroll 8
    for (int k = 0; k < H; ++k) acc = fmaf(hr[k], wr[k], acc);
    out[t] = acc;
}

extern "C" void kernel_launch(void* const* d_in, const int* in_sizes, int n_in,
                              void* d_out, int out_size, void* d_ws, size_t ws_size,
                              hipStream_t stream) {
    const float* x     = (const float*)d_in[0];
    const int*   erow  = (const int*)  d_in[1];
    const int*   ecol  = (const int*)  d_in[2];
    const float* ew    = (const float*)d_in[3];
    const float* W_in  = (const float*)d_in[4];
    const float* b_in  = (const float*)d_in[5];
    const float* W_out = (const float*)d_in[6];
    const float* b_out = (const float*)d_in[7];
    const float* W_cv  = (const float*)d_in[8];   // [L, H, H]

    // workspace: x0 | h0 | h1 | xx  (4 x N*H f32 = ~82 MB)
    float* x0 = (float*)d_ws;
    float* h0 = x0 + (size_t)N_NODES * H;
    float* h1 = h0 + (size_t)N_NODES * H;
    float* xx = h1 + (size_t)N_NODES * H;

    gcn_input_kernel<<<(N_NODES * H + 255) / 256, 256, 0, stream>>>(
        x, W_in, b_in, x0, h0);

    float* hc = h0;
    float* hn = h1;
    for (int l = 0; l < L_LAYERS; ++l) {
        float beta = logf(0.5f / (float)(l + 1) + 1.0f);
        gcn_init_xx<<<(N_NODES * H / 4 + 255) / 256, 256, 0, stream>>>(x0, xx);
        gcn_spmm<<<(N_EDGES * 32 + 255) / 256, 256, 0, stream>>>(
            erow, ecol, ew, hc, xx);
        gcn_layer_wmma<<<N_NODES / 16, 256, 0, stream>>>(
            xx, W_cv + (size_t)l * H * H, hc, hn, beta);
        float* tmp = hc; hc = hn; hn = tmp;
    }

    gcn_output_kernel<<<(N_NODES * C_OUT + 255) / 256, 256, 0, stream>>>(
        hc, W_out, b_out, (float*)d_out);
}